// VQCodebook_7490422964649
// MI455X (gfx1250) — compile-verified
//
#include <hip/hip_runtime.h>
#include <hip/hip_bf16.h>
#include <math.h>

#define B_SZ   8192
#define DLAT   512
#define NCB    2
#define KCODES 8192
#define DSUB   256

// d_out layout (floats, reference return order):
//   [0, 4194304)            quantized  [B, 512]
//   [4194304, 4210688)      indices    [B, 2]  (as float)
//   [4210688]               commitment_loss
//   [4210689]               codebook_loss (0)
//   [4210690]               perplexity
#define QUANT_OFF  0
#define IDX_OFF    (B_SZ * DLAT)
#define SCAL_OFF   (IDX_OFF + B_SZ * NCB)

// d_ws layout (32-bit words):
//   [0, 16384)      e_sq   float  [ncb, K]
//   [16384, 32768)  idx    int    [B, ncb]  flat b*2+n
//   [32768, 49152)  hist   uint   [ncb, K]
//   [49152]         loss accumulator (float)

typedef float v8f __attribute__((ext_vector_type(8)));
typedef float v2f __attribute__((ext_vector_type(2)));

#if defined(__HIP_DEVICE_COMPILE__)
#if !__has_builtin(__builtin_amdgcn_wmma_f32_16x16x4_f32)
#error "missing __builtin_amdgcn_wmma_f32_16x16x4_f32 on this toolchain"
#endif
#endif

// ---------------------------------------------------------------------------
// Kernel 1: e_sq per code row; also zero histogram and loss accumulator.
// ---------------------------------------------------------------------------
__global__ void vq_esq_init_kernel(const float* __restrict__ emb,
                                   float* __restrict__ esq,
                                   unsigned* __restrict__ hist,
                                   float* __restrict__ lossAcc) {
    int g = blockIdx.x * blockDim.x + threadIdx.x;   // [0, 16384)
    const float* row = emb + (size_t)g * DSUB;
    float s = 0.0f;
#pragma unroll 4
    for (int i = 0; i < DSUB; i += 4) {
        float4 v = *(const float4*)(row + i);
        s = fmaf(v.x, v.x, s);
        s = fmaf(v.y, v.y, s);
        s = fmaf(v.z, v.z, s);
        s = fmaf(v.w, v.w, s);
    }
    esq[g]  = s;
    hist[g] = 0u;
    if (g == 0) *lossAcc = 0.0f;
}

// ---------------------------------------------------------------------------
// Kernel 2: fused cross-GEMM (f32 WMMA 16x16x4) + argmin over K.
// grid = (256 M-tiles, 2 codebooks), block = 128 threads (4 waves).
// Block stages a 32-row z tile in LDS; each wave owns a 2048-wide N strip
// and accumulates 32(M)x64(N) per step: 2 A frags (LDS, reused x4) against
// 4 B frags (global) -> 8 WMMA per k-step, global:wmma = 1:2.
// ---------------------------------------------------------------------------
#define LDS_STRIDE 260   // 256 + 4 pad: bank = (4*row + kd) % 64, conflict-free

__global__ __launch_bounds__(128)
void vq_argmin_kernel(const float* __restrict__ z,
                      const float* __restrict__ emb,
                      const float* __restrict__ esq,
                      int* __restrict__ idxW,
                      unsigned* __restrict__ hist,
                      float* __restrict__ outIdxF) {
    __shared__ __align__(16) float sA[32 * LDS_STRIDE];
    __shared__ float sVal[4][32];
    __shared__ int   sIdx[4][32];

    const int n     = blockIdx.y;           // codebook
    const int rBase = blockIdx.x * 32;      // 32 rows of z
    const int tid   = threadIdx.x;

    // stage z tile [32 x 256] into LDS (2048 float4 / 128 threads)
    for (int i = tid; i < 32 * 64; i += 128) {
        int r  = i >> 6;
        int c4 = (i & 63) << 2;
        float4 v = *(const float4*)(z + (size_t)(rBase + r) * DLAT + n * DSUB + c4);
        *(float4*)(&sA[r * LDS_STRIDE + c4]) = v;
    }
    __syncthreads();

    const int wave = tid >> 5;
    const int lane = tid & 31;
    const int half = lane >> 4;     // 0: rows r, K=0..1 ; 1: rows r+8, K=2..3
    const int l16  = lane & 15;

    const float* eB   = emb + (size_t)n * KCODES * DSUB;
    const float* esqP = esq + (size_t)n * KCODES;

    float runVal[2][8];
    int   runIdx[2][8];
#pragma unroll
    for (int m = 0; m < 2; ++m)
#pragma unroll
        for (int r = 0; r < 8; ++r) { runVal[m][r] = 3.4e38f; runIdx[m][r] = 0; }

    const float* arow0 = &sA[l16 * LDS_STRIDE + (half << 1)];          // rows 0..15
    const float* arow1 = &sA[(16 + l16) * LDS_STRIDE + (half << 1)];   // rows 16..31

    const int nBeg = wave * (KCODES / 4);
    const int nEnd = nBeg + (KCODES / 4);

    for (int n0 = nBeg; n0 < nEnd; n0 += 64) {
        v8f acc[2][4];
#pragma unroll
        for (int m = 0; m < 2; ++m)
#pragma unroll
            for (int s = 0; s < 4; ++s) acc[m][s] = (v8f){};

        const float* ecol = eB + (size_t)(n0 + l16) * DSUB + (half << 1);
#pragma unroll 4
        for (int kd = 0; kd < DSUB; kd += 4) {
            v2f a0 = *(const v2f*)(arow0 + kd);
            v2f a1 = *(const v2f*)(arow1 + kd);
            v2f b0 = *(const v2f*)(ecol + kd);
            v2f b1 = *(const v2f*)(ecol + kd + 16 * DSUB);
            v2f b2 = *(const v2f*)(ecol + kd + 32 * DSUB);
            v2f b3 = *(const v2f*)(ecol + kd + 48 * DSUB);
            acc[0][0] = __builtin_amdgcn_wmma_f32_16x16x4_f32(false, a0, false, b0, (short)0, acc[0][0], false, false);
            acc[1][0] = __builtin_amdgcn_wmma_f32_16x16x4_f32(false, a1, false, b0, (short)0, acc[1][0], false, false);
            acc[0][1] = __builtin_amdgcn_wmma_f32_16x16x4_f32(false, a0, false, b1, (short)0, acc[0][1], false, false);
            acc[1][1] = __builtin_amdgcn_wmma_f32_16x16x4_f32(false, a1, false, b1, (short)0, acc[1][1], false, false);
            acc[0][2] = __builtin_amdgcn_wmma_f32_16x16x4_f32(false, a0, false, b2, (short)0, acc[0][2], false, false);
            acc[1][2] = __builtin_amdgcn_wmma_f32_16x16x4_f32(false, a1, false, b2, (short)0, acc[1][2], false, false);
            acc[0][3] = __builtin_amdgcn_wmma_f32_16x16x4_f32(false, a0, false, b3, (short)0, acc[0][3], false, false);
            acc[1][3] = __builtin_amdgcn_wmma_f32_16x16x4_f32(false, a1, false, b3, (short)0, acc[1][3], false, false);
        }
        // epilogue: score = e_sq[col] - 2*cross; running argmin (strict < keeps first index)
        const int colBase = n0 + l16;
#pragma unroll
        for (int s = 0; s < 4; ++s) {
            const int col = colBase + s * 16;
            const float es = esqP[col];
#pragma unroll
            for (int m = 0; m < 2; ++m) {
#pragma unroll
                for (int r = 0; r < 8; ++r) {
                    float sc = fmaf(-2.0f, acc[m][s][r], es);
                    bool upd = sc < runVal[m][r];
                    runVal[m][r] = upd ? sc  : runVal[m][r];
                    runIdx[m][r] = upd ? col : runIdx[m][r];
                }
            }
        }
    }

    // cross-lane min+index reduction within each 16-lane group
#pragma unroll
    for (int m = 0; m < 2; ++m) {
#pragma unroll
        for (int r = 0; r < 8; ++r) {
            float v = runVal[m][r];
            int   id = runIdx[m][r];
#pragma unroll
            for (int mk = 1; mk <= 8; mk <<= 1) {
                float ov = __shfl_xor(v, mk);
                int   oi = __shfl_xor(id, mk);
                if (ov < v || (ov == v && oi < id)) { v = ov; id = oi; }
            }
            if (l16 == 0) {
                sVal[wave][m * 16 + r + 8 * half] = v;
                sIdx[wave][m * 16 + r + 8 * half] = id;
            }
        }
    }
    __syncthreads();

    // combine the 4 waves' partial argmins (tie -> smaller index, matching jnp.argmin)
    if (tid < 32) {
        float v = sVal[0][tid];
        int  id = sIdx[0][tid];
#pragma unroll
        for (int w = 1; w < 4; ++w) {
            float ov = sVal[w][tid];
            int   oi = sIdx[w][tid];
            if (ov < v || (ov == v && oi < id)) { v = ov; id = oi; }
        }
        const int b = rBase + tid;
        idxW[b * NCB + n]    = id;
        outIdxF[b * NCB + n] = (float)id;
        atomicAdd(&hist[(size_t)n * KCODES + id], 1u);
    }
}

// ---------------------------------------------------------------------------
// Kernel 3: gather quantized vectors (STE output == z_q numerically) +
// commitment-loss partial sum. One block per batch row b, 128 threads.
// ---------------------------------------------------------------------------
__global__ __launch_bounds__(128)
void vq_gather_loss_kernel(const float* __restrict__ z,
                           const float* __restrict__ emb,
                           const int* __restrict__ idxW,
                           float* __restrict__ outQ,
                           float* __restrict__ lossAcc) {
    const int b = blockIdx.x;
    const int t = threadIdx.x;
    const int n  = t >> 6;            // codebook (64 threads each)
    const int d4 = (t & 63) << 2;     // float4 offset within sub-vector
    const int id = idxW[b * NCB + n];

    const float4 e4 = *(const float4*)(emb + ((size_t)n * KCODES + id) * DSUB + d4);
    const float4 z4 = *(const float4*)(z + (size_t)b * DLAT + n * DSUB + d4);
    *(float4*)(outQ + (size_t)b * DLAT + n * DSUB + d4) = e4;

    float dx = z4.x - e4.x, dy = z4.y - e4.y, dz = z4.z - e4.z, dw = z4.w - e4.w;
    float s = dx * dx + dy * dy + dz * dz + dw * dw;
#pragma unroll
    for (int m = 1; m < 32; m <<= 1) s += __shfl_xor(s, m);
    if ((t & 31) == 0) atomicAdd(lossAcc, s);
}

// ---------------------------------------------------------------------------
// Kernel 4: perplexity from histogram + scalar outputs.
// ---------------------------------------------------------------------------
__global__ void vq_finalize_kernel(const unsigned* __restrict__ hist,
                                   const float* __restrict__ lossAcc,
                                   float* __restrict__ out) {
    __shared__ float red[256];
    const int t = threadIdx.x;
    float s = 0.0f;
    for (int k = t; k < KCODES; k += 256) {
        float c = (float)hist[k] + (float)hist[KCODES + k];
        float p = c * (1.0f / (2.0f * (float)B_SZ));   // mean over ncb of counts/B
        s -= p * logf(p + 1e-10f);
    }
    red[t] = s;
    __syncthreads();
    for (int off = 128; off > 0; off >>= 1) {
        if (t < off) red[t] += red[t + off];
        __syncthreads();
    }
    if (t == 0) {
        out[SCAL_OFF + 0] = 0.25f * (*lossAcc) / (float)(B_SZ * DLAT);  // commitment
        out[SCAL_OFF + 1] = 0.0f;                                        // codebook
        out[SCAL_OFF + 2] = expf(red[0]);                                // perplexity
    }
}

// ---------------------------------------------------------------------------
extern "C" void kernel_launch(void* const* d_in, const int* in_sizes, int n_in,
                              void* d_out, int out_size, void* d_ws, size_t ws_size,
                              hipStream_t stream) {
    const float* z   = (const float*)d_in[0];   // [8192, 512]
    const float* emb = (const float*)d_in[1];   // [2, 8192, 256]
    float* out = (float*)d_out;

    float*    esq     = (float*)d_ws;                       // 16384 f32
    int*      idxW    = (int*)d_ws + 16384;                 // 16384 i32
    unsigned* hist    = (unsigned*)((int*)d_ws + 32768);    // 16384 u32
    float*    lossAcc = (float*)d_ws + 49152;               // 1 f32

    vq_esq_init_kernel<<<64, 256, 0, stream>>>(emb, esq, hist, lossAcc);

    dim3 grid(B_SZ / 32, NCB);   // 256 x 2
    vq_argmin_kernel<<<grid, 128, 0, stream>>>(z, emb, esq, idxW, hist,
                                               out + IDX_OFF);

    vq_gather_loss_kernel<<<B_SZ, 128, 0, stream>>>(z, emb, idxW,
                                                    out + QUANT_OFF, lossAcc);

    vq_finalize_kernel<<<1, 256, 0, stream>>>(hist, lossAcc, out);
}